// QuantumDenseNet_59631325937933
// MI455X (gfx1250) — compile-verified
//
#include <hip/hip_runtime.h>
#include <math.h>

// ---------------------------------------------------------------------------
// CV quantum circuit simulator for MI455X (gfx1250).
// C=6, N=5 modes, state = 6^5 = 7776 complex (SoA re/im planes, LDS resident).
// apply2 (80x per item, ~90% of FLOPs after folding) runs on
// v_wmma_f32_16x16x4_f32. All diagonal gates (rot/kerr) are algebraically
// folded into neighboring dense gates at gate-prep time (disjoint-mode
// commutation), so the per-item circuit is 125 dense passes: 80 WMMA apply2
// + 45 VALU apply1.
// Assumptions (uncheckable at runtime here):
//  * B-fragment (4x16 f32) layout mirrors A: lane holds B[K=2*(lane/16)+v][N=lane%16]
//  * C/D layout per ISA: VGPR v holds row M=v (+8 for lanes 16-31), col N=lane%16
//  * ws_size >= ~2.3 MB
// ---------------------------------------------------------------------------

typedef __attribute__((ext_vector_type(2))) float v2f;
typedef __attribute__((ext_vector_type(8))) float v8f;

#define GAS __attribute__((address_space(1)))

// force global (not flat) vector-memory path for gate reads
__device__ __forceinline__ v2f ldg_v2(const float* p) {
  return *(const GAS v2f*)(uintptr_t)p;
}
__device__ __forceinline__ float ldg_f(const float* p) {
  return *(const GAS float*)(uintptr_t)p;
}

constexpr int P6[6] = {1, 6, 36, 216, 1296, 7776};
constexpr int NT = 256;                     // threads per block in k_main

// workspace layout, units of float. Gates stored SPLIT (re plane, im plane).
constexpr int BSF    = 2592;                // per bs gate: Ur[1296] then Ui[1296]
constexpr int SQ_OFF = 80 * BSF;            // 20 squeeze gates: real 36
constexpr int DI_OFF = SQ_OFF + 20 * 36;    // 20 disp gates (kerr folded): re36,im36
constexpr int XD_OFF = DI_OFF + 20 * 72;    // batch disp gates: real 36 each

// ---------------------------------------------------------------------------
// small dense expm helpers (order 16, s=6 scaling-and-squaring, matches ref)
// ---------------------------------------------------------------------------
__device__ __forceinline__ void matmul6(const float* A, const float* B, float* Cc) {
#pragma unroll
  for (int i = 0; i < 6; ++i)
#pragma unroll
    for (int j = 0; j < 6; ++j) {
      float acc = 0.f;
#pragma unroll
      for (int k = 0; k < 6; ++k) acc += A[i*6+k] * B[k*6+j];
      Cc[i*6+j] = acc;
    }
}

__device__ void expm_r6(const float* H, float* R) {
  float Hs[36], T[36];
#pragma unroll
  for (int i = 0; i < 36; ++i) { Hs[i] = H[i] * (1.f/64.f); R[i] = (i % 7 == 0) ? 1.f : 0.f; }
  for (int k = 16; k >= 1; --k) {
    matmul6(Hs, R, T);
    float invk = 1.f / (float)k;
#pragma unroll
    for (int i = 0; i < 36; ++i) R[i] = ((i % 7 == 0) ? 1.f : 0.f) + T[i] * invk;
  }
  for (int s = 0; s < 6; ++s) {
    matmul6(R, R, T);
#pragma unroll
    for (int i = 0; i < 36; ++i) R[i] = T[i];
  }
}

__device__ __forceinline__ void cmatmul6(const float2* A, const float2* B, float2* Cc) {
#pragma unroll
  for (int i = 0; i < 6; ++i)
#pragma unroll
    for (int j = 0; j < 6; ++j) {
      float ar = 0.f, ai = 0.f;
#pragma unroll
      for (int k = 0; k < 6; ++k) {
        float2 a = A[i*6+k], b = B[k*6+j];
        ar += a.x*b.x - a.y*b.y;
        ai += a.x*b.y + a.y*b.x;
      }
      Cc[i*6+j] = make_float2(ar, ai);
    }
}

__device__ void expm_c6(const float2* H, float2* R) {
  float2 Hs[36], T[36];
#pragma unroll
  for (int i = 0; i < 36; ++i) {
    Hs[i] = make_float2(H[i].x*(1.f/64.f), H[i].y*(1.f/64.f));
    R[i]  = make_float2((i % 7 == 0) ? 1.f : 0.f, 0.f);
  }
  for (int k = 16; k >= 1; --k) {
    cmatmul6(Hs, R, T);
    float invk = 1.f / (float)k;
#pragma unroll
    for (int i = 0; i < 36; ++i)
      R[i] = make_float2(((i % 7 == 0) ? 1.f : 0.f) + T[i].x*invk, T[i].y*invk);
  }
  for (int s = 0; s < 6; ++s) {
    cmatmul6(R, R, T);
#pragma unroll
    for (int i = 0; i < 36; ++i) R[i] = T[i];
  }
}

// ---------------------------------------------------------------------------
// Kernel 1: 80 beamsplitter gates. H = ep*kron(A,A^T) - conj(ep)*kron(A^T,A),
// expm(theta*H) via LDS 36x36 complex matmuls. Rot diagonals of the same
// interferometer are folded as a row scaling into gates n=8 (modes 0,1) and
// n=9 (modes 2,3): rot commutes with bs gates on disjoint modes.
// Output split re/im planes.
// ---------------------------------------------------------------------------
__global__ void __launch_bounds__(256) k_bs_gates(const float* __restrict__ lin,
                                                  float* __restrict__ ws) {
  __shared__ float2 Hs[1296];
  __shared__ float2 Rm[1296];
  __shared__ float2 Tm[1296];
  int blk = blockIdx.x;           // 0..79
  int l = blk / 20;
  int inter = (blk / 10) % 2;
  int n = blk % 10;
  const float* p = lin + l*63 + (inter ? 29 : 0);
  float theta = p[n];
  float phi   = p[10 + n];
  float epr = cosf(phi), epi = sinf(phi);
  float sc = theta * (1.f/64.f);

  for (int idx = threadIdx.x; idx < 1296; idx += 256) {
    int pp = idx / 36, q = idx % 36;
    int a = pp / 6, bb = pp % 6, c = q / 6, d = q % 6;
    float re = 0.f, im = 0.f;
    if (c == a + 1 && d == bb - 1) {            // ep * kron(A, A^T)
      float w = sqrtf((float)((a + 1) * bb));
      re += epr * w; im += epi * w;
    }
    if (c == a - 1 && d == bb + 1) {            // - conj(ep) * kron(A^T, A)
      float w = sqrtf((float)(a * (bb + 1)));
      re -= epr * w; im += epi * w;
    }
    Hs[idx] = make_float2(re * sc, im * sc);
    Rm[idx] = make_float2((pp == q) ? 1.f : 0.f, 0.f);
  }
  __syncthreads();

  for (int k = 16; k >= 1; --k) {
    for (int idx = threadIdx.x; idx < 1296; idx += 256) {
      int r = idx / 36, cidx = idx % 36;
      float ar = 0.f, ai = 0.f;
      for (int e = 0; e < 36; ++e) {
        float2 h = Hs[r*36 + e]; float2 xx = Rm[e*36 + cidx];
        ar += h.x*xx.x - h.y*xx.y;
        ai += h.x*xx.y + h.y*xx.x;
      }
      Tm[idx] = make_float2(ar, ai);
    }
    __syncthreads();
    float invk = 1.f / (float)k;
    for (int idx = threadIdx.x; idx < 1296; idx += 256) {
      int r = idx / 36, cidx = idx % 36;
      Rm[idx] = make_float2(((r == cidx) ? 1.f : 0.f) + Tm[idx].x*invk, Tm[idx].y*invk);
    }
    __syncthreads();
  }
  for (int s = 0; s < 6; ++s) {
    for (int idx = threadIdx.x; idx < 1296; idx += 256) {
      int r = idx / 36, cidx = idx % 36;
      float ar = 0.f, ai = 0.f;
      for (int e = 0; e < 36; ++e) {
        float2 u = Rm[r*36 + e]; float2 v = Rm[e*36 + cidx];
        ar += u.x*v.x - u.y*v.y;
        ai += u.x*v.y + u.y*v.x;
      }
      Tm[idx] = make_float2(ar, ai);
    }
    __syncthreads();
    for (int idx = threadIdx.x; idx < 1296; idx += 256) Rm[idx] = Tm[idx];
    __syncthreads();
  }

  // rot fold: rows p=(a,b) of gate n=8 scaled by exp(i(r0*a + r1*b)),
  // of gate n=9 by exp(i(r2*a + r3*b)).
  float rA = 0.f, rB = 0.f;
  bool fold = false;
  if (n == 8) { rA = p[20 + 0]; rB = p[20 + 1]; fold = true; }
  if (n == 9) { rA = p[20 + 2]; rB = p[20 + 3]; fold = true; }

  float* g = ws + (size_t)blk * BSF;
  for (int idx = threadIdx.x; idx < 1296; idx += 256) {
    float2 v = Rm[idx];
    if (fold) {
      int pp = idx / 36; int a = pp / 6, bb = pp % 6;
      float ang = rA * (float)a + rB * (float)bb;
      float cc = cosf(ang), ss = sinf(ang);
      v = make_float2(cc*v.x - ss*v.y, cc*v.y + ss*v.x);
    }
    g[idx]        = v.x;                        // Ur plane
    g[1296 + idx] = v.y;                        // Ui plane
  }
}

// ---------------------------------------------------------------------------
// Kernel 2: squeeze matrices (real) + layer-disp matrices with the following
// kerr diagonal folded in (kerr commutes with disp gates on other modes).
// ---------------------------------------------------------------------------
__global__ void __launch_bounds__(64) k_small_gates(const float* __restrict__ lin,
                                                    const float* __restrict__ act,
                                                    const float* __restrict__ lact,
                                                    float* __restrict__ ws) {
  int tid = threadIdx.x;
  if (tid < 20) {                               // squeeze: real expm
    int l = tid / 5, i = tid % 5;
    float r = lin[l*63 + 24 + i];
    float H[36];
#pragma unroll
    for (int k = 0; k < 36; ++k) H[k] = 0.f;
#pragma unroll
    for (int j = 0; j < 4; ++j) {
      float w = 0.5f * r * sqrtf((float)((j + 1) * (j + 2)));
      H[j*6 + (j + 2)] = w;
      H[(j + 2)*6 + j] = -w;
    }
    float R[36]; expm_r6(H, R);
    float* o = ws + SQ_OFF + (size_t)(l*5 + i) * 36;
#pragma unroll
    for (int k = 0; k < 36; ++k) o[k] = R[k];
  } else if (tid >= 32 && tid < 52) {           // disp (complex expm) + kerr fold
    int t = tid - 32; int l = t / 5, i = t % 5;
    float dr = lin[l*63 + 53 + i], dp = lin[l*63 + 58 + i];
    float alr = dr * cosf(dp), ali = dr * sinf(dp);
    float2 H[36];
#pragma unroll
    for (int k = 0; k < 36; ++k) H[k] = make_float2(0.f, 0.f);
#pragma unroll
    for (int j = 0; j < 5; ++j) {
      float w = sqrtf((float)(j + 1));
      H[(j + 1)*6 + j] = make_float2(alr * w, ali * w);    //  al * AD
      H[j*6 + (j + 1)] = make_float2(-alr * w, ali * w);   // -conj(al) * A
    }
    float2 R[36]; expm_c6(H, R);
    // kerr fold: layers 0..2 use act[l,i]; layer 3 uses last_act[i] for i<2
    float kap = (l < 3) ? act[l*5 + i] : ((i < 2) ? lact[i] : 0.f);
    float* o = ws + DI_OFF + (size_t)t * 72;
#pragma unroll
    for (int a = 0; a < 6; ++a) {
      float ang = kap * (float)(a * a);
      float kc = cosf(ang), ks = sinf(ang);
#pragma unroll
      for (int b = 0; b < 6; ++b) {
        float2 v = R[a*6 + b];
        o[a*6 + b]      = kc*v.x - ks*v.y;      // re plane
        o[36 + a*6 + b] = kc*v.y + ks*v.x;      // im plane
      }
    }
  }
}

// ---------------------------------------------------------------------------
// Kernel 3: batch initial displacement gates, disp(r,0)=expm(r(AD-A)) (real).
// ---------------------------------------------------------------------------
__global__ void __launch_bounds__(256) k_batch_disp(const float* __restrict__ x,
                                                    float* __restrict__ ws, int total) {
  int g = blockIdx.x * blockDim.x + threadIdx.x;
  if (g >= total) return;
  float r = x[g];
  float H[36];
#pragma unroll
  for (int k = 0; k < 36; ++k) H[k] = 0.f;
#pragma unroll
  for (int j = 0; j < 5; ++j) {
    float w = r * sqrtf((float)(j + 1));
    H[(j + 1)*6 + j] = w;
    H[j*6 + (j + 1)] = -w;
  }
  float R[36]; expm_r6(H, R);
  float* o = ws + XD_OFF + (size_t)g * 36;
#pragma unroll
  for (int k = 0; k < 36; ++k) o[k] = R[k];
}

// ---------------------------------------------------------------------------
// Main-kernel gate primitives (psi SoA in LDS, in-place, barrier per gate).
// ---------------------------------------------------------------------------
template <int M>   // real 6x6 gate
__device__ __forceinline__ void apply1r(const float* __restrict__ U,
                                        float* pre, float* pim, int tid) {
  constexpr int s = P6[4 - M];
  float u[36];
#pragma unroll
  for (int k = 0; k < 36; ++k) u[k] = ldg_f(U + k);
  for (int f = tid; f < 1296; f += NT) {
    int outer = f / s, inner = f % s;
    int base = outer * 6 * s + inner;
    float vr[6], vi[6];
#pragma unroll
    for (int b = 0; b < 6; ++b) { vr[b] = pre[base + b*s]; vi[b] = pim[base + b*s]; }
#pragma unroll
    for (int a = 0; a < 6; ++a) {
      float ar = 0.f, ai = 0.f;
#pragma unroll
      for (int b = 0; b < 6; ++b) { ar += u[a*6+b]*vr[b]; ai += u[a*6+b]*vi[b]; }
      pre[base + a*s] = ar; pim[base + a*s] = ai;
    }
  }
  __syncthreads();
}

template <int M>   // complex 6x6 gate, split storage (Ure[36], Uim[36])
__device__ __forceinline__ void apply1c(const float* __restrict__ U,
                                        float* pre, float* pim, int tid) {
  constexpr int s = P6[4 - M];
  float ur[36], ui[36];
#pragma unroll
  for (int k = 0; k < 36; ++k) { ur[k] = ldg_f(U + k); ui[k] = ldg_f(U + 36 + k); }
  for (int f = tid; f < 1296; f += NT) {
    int outer = f / s, inner = f % s;
    int base = outer * 6 * s + inner;
    float vr[6], vi[6];
#pragma unroll
    for (int b = 0; b < 6; ++b) { vr[b] = pre[base + b*s]; vi[b] = pim[base + b*s]; }
#pragma unroll
    for (int a = 0; a < 6; ++a) {
      float ar = 0.f, ai = 0.f;
#pragma unroll
      for (int b = 0; b < 6; ++b) {
        ar += ur[a*6+b]*vr[b] - ui[a*6+b]*vi[b];
        ai += ur[a*6+b]*vi[b] + ui[a*6+b]*vr[b];
      }
      pre[base + a*s] = ar; pim[base + a*s] = ai;
    }
  }
  __syncthreads();
}

// apply2: out = U(36x36 complex) x in(36x216 fibers) via v_wmma_f32_16x16x4_f32.
// One wave owns a 16-fiber column group (all 36 rows) -> in-place safe.
// re-acc P gets Ur*Xr + Ui*(-Xi); im-acc R gets Ur*Xi + Ui*Xr.
template <int M2>
__device__ void apply2w(const float* __restrict__ Ug, float* pre, float* pim, int tid) {
  constexpr int s2 = P6[3 - M2];
  int wave = tid >> 5;
  int lane = tid & 31;
  int lr = lane & 15;        // row/col within 16
  int lh = lane >> 4;        // lane half selects K group
  for (int j = wave; j < 14; j += 8) {          // 14 column tiles of 16 fibers
    // launder gate pointer so A-fragment loads are not hoisted across j
    // (keeps VGPR pressure < 256; loads are L1/L2-hot, shared by all blocks)
    uintptr_t ul = (uintptr_t)Ug;
    asm volatile("" : "+s"(ul));
    const float* ur = (const float*)ul;
    const float* ui = ur + 1296;

    int f = j * 16 + lr;
    int fc = (f < 216) ? f : 215;               // clamp: garbage cols never stored
    int outer = fc / s2, inner = fc % s2;
    int base = outer * 36 * s2 + inner;
    v8f P[3] = {}; v8f Rr[3] = {};
    for (int kt = 0; kt < 9; ++kt) {            // K = 36 = 9 x 4
      int q0 = kt * 4 + 2 * lh;
      v2f bR, bI;
      bR[0] = pre[base + q0 * s2];
      bR[1] = pre[base + (q0 + 1) * s2];
      bI[0] = pim[base + q0 * s2];
      bI[1] = pim[base + (q0 + 1) * s2];
      v2f bIn = -bI;                            // shared negation (no f32 A/B neg mod)
#pragma unroll
      for (int i = 0; i < 3; ++i) {             // 3 row tiles (rows 0..47, 36 valid)
        int r = i * 16 + lr;
        int rc = (r < 36) ? r : 35;             // clamp: garbage rows never stored
        v2f aR = ldg_v2(ur + rc * 36 + q0);     // 8B-aligned global_load_b64
        v2f aI = ldg_v2(ui + rc * 36 + q0);
        P[i]  = __builtin_amdgcn_wmma_f32_16x16x4_f32(false, aR, false, bR,  (short)0, P[i],  false, false);
        P[i]  = __builtin_amdgcn_wmma_f32_16x16x4_f32(false, aI, false, bIn, (short)0, P[i],  false, false);
        Rr[i] = __builtin_amdgcn_wmma_f32_16x16x4_f32(false, aR, false, bI,  (short)0, Rr[i], false, false);
        Rr[i] = __builtin_amdgcn_wmma_f32_16x16x4_f32(false, aI, false, bR,  (short)0, Rr[i], false, false);
      }
    }
    if (f < 216) {
#pragma unroll
      for (int i = 0; i < 3; ++i)
#pragma unroll
        for (int v = 0; v < 8; ++v) {
          int p = i * 16 + 8 * lh + v;          // D row (ISA C/D layout)
          if (p < 36) {
            pre[base + p * s2] = P[i][v];
            pim[base + p * s2] = Rr[i][v];
          }
        }
    }
  }
  __syncthreads();
}

__device__ __forceinline__ void do_a2(const float* U, float* pre, float* pim, int k, int tid) {
  switch (k) {
    case 0: apply2w<0>(U, pre, pim, tid); break;
    case 1: apply2w<1>(U, pre, pim, tid); break;
    case 2: apply2w<2>(U, pre, pim, tid); break;
    default: apply2w<3>(U, pre, pim, tid); break;
  }
}
__device__ __forceinline__ void do_a1r(const float* U, float* pre, float* pim, int m, int tid) {
  switch (m) {
    case 0: apply1r<0>(U, pre, pim, tid); break;
    case 1: apply1r<1>(U, pre, pim, tid); break;
    case 2: apply1r<2>(U, pre, pim, tid); break;
    case 3: apply1r<3>(U, pre, pim, tid); break;
    default: apply1r<4>(U, pre, pim, tid); break;
  }
}
__device__ __forceinline__ void do_a1c(const float* U, float* pre, float* pim, int m, int tid) {
  switch (m) {
    case 0: apply1c<0>(U, pre, pim, tid); break;
    case 1: apply1c<1>(U, pre, pim, tid); break;
    case 2: apply1c<2>(U, pre, pim, tid); break;
    case 3: apply1c<3>(U, pre, pim, tid); break;
    default: apply1c<4>(U, pre, pim, tid); break;
  }
}

// ---------------------------------------------------------------------------
// Kernel 4: full circuit, one workgroup (8 wave32) per batch item.
// Sequence per layer (rot/kerr pre-folded): 10 bs, 5 squeeze, 10 bs, 5 disp.
// ---------------------------------------------------------------------------
__global__ void __launch_bounds__(NT) k_main(const float* __restrict__ ws,
                                             float* __restrict__ out) {
  __shared__ float psiRe[7776];               // SoA state, 62.2 KB total
  __shared__ float psiIm[7776];
  __shared__ float red[NT];
  int b = blockIdx.x;
  int tid = threadIdx.x;

  for (int idx = tid; idx < 7776; idx += NT) {
    psiRe[idx] = (idx == 0) ? 1.f : 0.f;
    psiIm[idx] = 0.f;
  }
  __syncthreads();

  // initial displacements (batch-dependent, real matrices)
  for (int i = 0; i < 5; ++i)
    do_a1r(ws + XD_OFF + (size_t)(b * 5 + i) * 36, psiRe, psiIm, i, tid);

  const int BSK[10] = {0, 2, 1, 3, 0, 2, 1, 3, 0, 2};  // bs mode sequence
  for (int l = 0; l < 4; ++l) {
    for (int half = 0; half < 2; ++half) {
      for (int n = 0; n < 10; ++n)
        do_a2(ws + (size_t)(l * 20 + half * 10 + n) * BSF, psiRe, psiIm, BSK[n], tid);
      if (half == 0)
        for (int i = 0; i < 5; ++i)
          do_a1r(ws + SQ_OFF + (size_t)(l * 5 + i) * 36, psiRe, psiIm, i, tid);
      else
        for (int i = 0; i < 5; ++i)
          do_a1c(ws + DI_OFF + (size_t)(l * 5 + i) * 72, psiRe, psiIm, i, tid);
    }
  }

  // <X_i> = 2 * sum_fibers sum_a sqrt(a+1) Re(conj(psi_a) psi_{a+1})
  for (int i = 0; i < 2; ++i) {
    int s = (i == 0) ? 1296 : 216;
    float part = 0.f;
    for (int f = tid; f < 1296; f += NT) {
      int outer = f / s, inner = f % s;
      int base = outer * 6 * s + inner;
      float vr[6], vi[6];
#pragma unroll
      for (int a = 0; a < 6; ++a) { vr[a] = psiRe[base + a*s]; vi[a] = psiIm[base + a*s]; }
#pragma unroll
      for (int a = 0; a < 5; ++a)
        part += sqrtf((float)(a + 1)) * (vr[a]*vr[a+1] + vi[a]*vi[a+1]);
    }
    red[tid] = part;
    __syncthreads();
    for (int off = NT / 2; off > 0; off >>= 1) {
      if (tid < off) red[tid] += red[tid + off];
      __syncthreads();
    }
    if (tid == 0) out[b * 2 + i] = 2.f * red[0];
    __syncthreads();
  }
}

// ---------------------------------------------------------------------------
extern "C" void kernel_launch(void* const* d_in, const int* in_sizes, int n_in,
                              void* d_out, int out_size, void* d_ws, size_t ws_size,
                              hipStream_t stream) {
  const float* x    = (const float*)d_in[0];   // (B,5)
  const float* lin  = (const float*)d_in[1];   // (4,63)
  const float* act  = (const float*)d_in[2];   // (3,5)
  const float* lact = (const float*)d_in[3];   // (5,)
  float* out = (float*)d_out;                  // (B,2)
  float* ws  = (float*)d_ws;

  int B = in_sizes[0] / 5;
  int total = B * 5;

  k_bs_gates<<<80, 256, 0, stream>>>(lin, ws);
  k_small_gates<<<1, 64, 0, stream>>>(lin, act, lact, ws);
  k_batch_disp<<<(total + 255) / 256, 256, 0, stream>>>(x, ws, total);
  k_main<<<B, 256, 0, stream>>>(ws, out);
  (void)n_in; (void)out_size; (void)ws_size;
}